// ArmModel_34943853920533
// MI455X (gfx1250) — compile-verified
//
#include <hip/hip_runtime.h>
#include <math.h>

#define DOF 7
#define BATCH 8192
#define NT 128
#define DT 0.1f
#define ACTION_RANGE 50.0f
#define MAX_VEL 20.0f

typedef __attribute__((ext_vector_type(2))) float v2f;
typedef __attribute__((ext_vector_type(8))) float v8f;

__device__ __forceinline__ void mat3mul(const float* A, const float* B, float* C) {
#pragma unroll
  for (int r = 0; r < 3; ++r)
#pragma unroll
    for (int c = 0; c < 3; ++c)
      C[r*3+c] = A[r*3+0]*B[0*3+c] + A[r*3+1]*B[1*3+c] + A[r*3+2]*B[2*3+c];
}
__device__ __forceinline__ void mat3vec(const float* A, const float* x, float* y) {
  y[0]=A[0]*x[0]+A[1]*x[1]+A[2]*x[2];
  y[1]=A[3]*x[0]+A[4]*x[1]+A[5]*x[2];
  y[2]=A[6]*x[0]+A[7]*x[1]+A[8]*x[2];
}
__device__ __forceinline__ void mat3Tvec(const float* A, const float* x, float* y) {
  y[0]=A[0]*x[0]+A[3]*x[1]+A[6]*x[2];
  y[1]=A[1]*x[0]+A[4]*x[1]+A[7]*x[2];
  y[2]=A[2]*x[0]+A[5]*x[1]+A[8]*x[2];
}
__device__ __forceinline__ void cross3(const float* a, const float* b, float* c) {
  c[0]=a[1]*b[2]-a[2]*b[1];
  c[1]=a[2]*b[0]-a[0]*b[2];
  c[2]=a[0]*b[1]-a[1]*b[0];
}

// exp of se3 twist (w,v) -> rotation R (3x3) and translation p, matching the
// reference's small-angle series switch at |w| < 1e-4.
__device__ __forceinline__ void exp_se3(const float* w, const float* v, float* R, float* p) {
  float th2 = w[0]*w[0]+w[1]*w[1]+w[2]*w[2];
  float th  = sqrtf(th2);
  float a, b, c;
  if (th < 1e-4f) {
    a = 1.f - th2*(1.f/6.f);
    b = 0.5f - th2*(1.f/24.f);
    c = (1.f/6.f) - th2*(1.f/120.f);
  } else {
    float s = sinf(th), co = cosf(th);
    a = s/th; b = (1.f-co)/th2; c = (th-s)/(th2*th);
  }
  // K^2 = w w^T - th2 * I
  R[0]=1.f + b*(w[0]*w[0]-th2); R[1]=-a*w[2]+b*w[0]*w[1]; R[2]= a*w[1]+b*w[0]*w[2];
  R[3]= a*w[2]+b*w[0]*w[1];     R[4]=1.f + b*(w[1]*w[1]-th2); R[5]=-a*w[0]+b*w[1]*w[2];
  R[6]=-a*w[1]+b*w[0]*w[2];     R[7]= a*w[0]+b*w[1]*w[2]; R[8]=1.f + b*(w[2]*w[2]-th2);
  float Vm[9];
  Vm[0]=1.f + c*(w[0]*w[0]-th2); Vm[1]=-b*w[2]+c*w[0]*w[1]; Vm[2]= b*w[1]+c*w[0]*w[2];
  Vm[3]= b*w[2]+c*w[0]*w[1];     Vm[4]=1.f + c*(w[1]*w[1]-th2); Vm[5]=-b*w[0]+c*w[1]*w[2];
  Vm[6]=-b*w[1]+c*w[0]*w[2];     Vm[7]= b*w[0]+c*w[1]*w[2]; Vm[8]=1.f + c*(w[2]*w[2]-th2);
  mat3vec(Vm, v, p);
}

// out = Adjoint([R p;0 1]) * in   (in/out are 6-vectors (w,v))
__device__ __forceinline__ void adTapply(const float* Tr, const float* Tp,
                                         const float* in, float* out) {
  float wo[3], rv[3], cp[3];
  mat3vec(Tr, in, wo);
  mat3vec(Tr, in+3, rv);
  cross3(Tp, wo, cp);
  out[0]=wo[0]; out[1]=wo[1]; out[2]=wo[2];
  out[3]=cp[0]+rv[0]; out[4]=cp[1]+rv[1]; out[5]=cp[2]+rv[2];
}
// out = Adjoint([R p;0 1])^T * F  : out_m = R^T (Fm - p x Ff), out_f = R^T Ff
__device__ __forceinline__ void adTTapply(const float* Tr, const float* Tp,
                                          const float* F, float* out) {
  float cp[3], t[3], om[3], of[3];
  cross3(Tp, F+3, cp);
  t[0]=F[0]-cp[0]; t[1]=F[1]-cp[1]; t[2]=F[2]-cp[2];
  mat3Tvec(Tr, t, om);
  mat3Tvec(Tr, F+3, of);
  out[0]=om[0]; out[1]=om[1]; out[2]=om[2];
  out[3]=of[0]; out[4]=of[1]; out[5]=of[2];
}
__device__ __forceinline__ void g6mv(const float* G, const float* x, float* y) {
#pragma unroll
  for (int r=0;r<6;++r){
    float s=0.f;
#pragma unroll
    for (int c=0;c<6;++c) s += G[r*6+c]*x[c];
    y[r]=s;
  }
}
__device__ __forceinline__ float dot6(const float* a, const float* b){
  return a[0]*b[0]+a[1]*b[1]+a[2]*b[2]+a[3]*b[3]+a[4]*b[4]+a[5]*b[5];
}

__global__ __launch_bounds__(NT)
void arm_rk4_kernel(const float* __restrict__ state, const float* __restrict__ torque,
                    const float* __restrict__ Mg, const float* __restrict__ Ag,
                    const float* __restrict__ Lg, const float* __restrict__ gg,
                    const float* __restrict__ fg,
                    float* __restrict__ out_state, float* __restrict__ out_ee)
{
  __shared__ float sA[DOF][6];        // screw axes
  __shared__ float sG[DOF][36];       // G = L L^T (computed via WMMA)
  __shared__ float sCr[DOF+1][9];     // inv_T(M_i): R^T
  __shared__ float sCp[DOF+1][3];     // inv_T(M_i): -R^T p
  __shared__ float sMr[DOF+1][9];     // M_i rotation (FK)
  __shared__ float sMp[DOF+1][3];     // M_i translation (FK)
  __shared__ float sg[3];
  __shared__ float sft[6];
  __shared__ float trp[NT][85];       // per-thread 7 x (R 9 + p 3), stride 85 (odd -> no bank conflict)

  const int t = threadIdx.x;

  if (t < 42) ((float*)sA)[t] = Ag[t];
  if (t < 3)  sg[t] = gg[t];
  if (t < 6)  sft[t] = fg[t];
  if (t < DOF+1) {
    const float* Mi = Mg + t*16;   // row-major 4x4
#pragma unroll
    for (int r=0;r<3;++r){
#pragma unroll
      for (int c=0;c<3;++c) sMr[t][r*3+c] = Mi[r*4+c];
      sMp[t][r] = Mi[r*4+3];
    }
#pragma unroll
    for (int r=0;r<3;++r){
#pragma unroll
      for (int c=0;c<3;++c) sCr[t][r*3+c] = Mi[c*4+r];           // R^T
      sCp[t][r] = -(Mi[0*4+r]*Mi[0*4+3] + Mi[1*4+r]*Mi[1*4+3] + Mi[2*4+r]*Mi[2*4+3]);
    }
  }

  // ---- G_i = L_i L_i^T via V_WMMA_F32_16X16X4_F32 (exact f32 FMA) ----
  // A-operand (16x4): lane l -> m=l&15, VGPR j -> k=j+2*(l>>4); value L[m][k].
  // For L*L^T the B-operand (k,n)->L[n][k] has the *same* lane/VGPR mapping,
  // so A and B registers are identical. Two chained WMMAs cover K=0..7 (6 real).
  {
    const int wv   = t >> 5;
    const int lane = t & 31;
    const int half = lane >> 4;
    const int mr   = lane & 15;
    for (int i = wv; i < DOF; i += NT/32) {   // uniform per wave -> EXEC all-ones inside
      const float* Li = Lg + i*36;
      v2f alo, ahi;
#pragma unroll
      for (int j=0;j<2;++j){
        int k0 = j + 2*half;          // 0..3
        int k1 = 4 + j + 2*half;      // 4..7
        alo[j] = (mr < 6)            ? Li[mr*6+k0] : 0.f;
        ahi[j] = (mr < 6 && k1 < 6)  ? Li[mr*6+k1] : 0.f;
      }
      v8f c = {0.f,0.f,0.f,0.f,0.f,0.f,0.f,0.f};
      c = __builtin_amdgcn_wmma_f32_16x16x4_f32(false, alo, false, alo, (short)0, c, false, false);
      c = __builtin_amdgcn_wmma_f32_16x16x4_f32(false, ahi, false, ahi, (short)0, c, false, false);
#pragma unroll
      for (int r=0;r<8;++r){
        int m = r + 8*half;
        if (m < 6 && mr < 6) sG[i][m*6+mr] = c[r];
      }
    }
  }
  __syncthreads();

  const int idx = blockIdx.x * NT + t;

  float q0[7], dq0[7], tf[7];
#pragma unroll
  for (int j=0;j<7;++j){
    q0[j]  = state[idx*14 + j];
    dq0[j] = state[idx*14 + 7 + j];
    tf[j]  = torque[idx*7 + j] * ACTION_RANGE;
  }

  float qs[7], dqs[7], accq[7], accdq[7];
#pragma unroll
  for (int j=0;j<7;++j){ qs[j]=q0[j]; dqs[j]=dq0[j]; accq[j]=0.f; accdq[j]=0.f; }

  float* myT = trp[t];

#pragma unroll 1
  for (int st=0; st<4; ++st) {
    // ---- per-joint transforms T_i = exp(-A_i q_i) * inv_T(M_i), stored (R,p) in LDS ----
#pragma unroll
    for (int i=0;i<7;++i){
      float w[3], v[3], Er[9], Ep[3];
#pragma unroll
      for (int u=0;u<3;++u){ w[u] = -sA[i][u]*qs[i]; v[u] = -sA[i][3+u]*qs[i]; }
      exp_se3(w, v, Er, Ep);
      float tmp9[9]; mat3mul(Er, sCr[i], tmp9);
      float tp3[3];  mat3vec(Er, sCp[i], tp3);
      float* Tr = myT + i*12;
#pragma unroll
      for (int u=0;u<9;++u) Tr[u] = tmp9[u];
#pragma unroll
      for (int u=0;u<3;++u) Tr[9+u] = tp3[u] + Ep[u];
    }

    // ---- h = InverseDynamics(q, dq, ddq=0, g, Ftip) ----
    float V[6]  = {0,0,0,0,0,0};
    float Vd[6] = {0,0,0,-sg[0],-sg[1],-sg[2]};
    float Vsave[7][6], Vdsave[7][6];
#pragma unroll
    for (int i=0;i<7;++i){
      const float* Tr = myT + i*12;
      const float* Tp = Tr + 9;
      float tv[6]; adTapply(Tr, Tp, V, tv);
#pragma unroll
      for (int u=0;u<6;++u) V[u] = tv[u] + sA[i][u]*dqs[i];
      float tvd[6]; adTapply(Tr, Tp, Vd, tvd);
      float c1[3], c2[3], c3[3];                 // ad(V) @ A_i
      cross3(V,   sA[i],   c1);                  // w x Aw
      cross3(V+3, sA[i],   c2);                  // v x Aw
      cross3(V,   sA[i]+3, c3);                  // w x Av
#pragma unroll
      for (int u=0;u<3;++u){
        Vd[u]   = tvd[u]   + c1[u]*dqs[i];
        Vd[3+u] = tvd[3+u] + (c2[u]+c3[u])*dqs[i];
      }
#pragma unroll
      for (int u=0;u<6;++u){ Vsave[i][u]=V[u]; Vdsave[i][u]=Vd[u]; }
    }
    float F[6];
#pragma unroll
    for (int u=0;u<6;++u) F[u] = sft[u];
    float h[7];
#pragma unroll
    for (int i=6;i>=0;--i){
      float Fn[6];
      if (i==6) adTTapply(sCr[7], sCp[7], F, Fn);                  // Ad(inv_T(M7))^T
      else      adTTapply(myT + (i+1)*12, myT + (i+1)*12 + 9, F, Fn);
      float U[6], W[6];
      g6mv(sG[i], Vdsave[i], U);
      g6mv(sG[i], Vsave[i],  W);
      float cm1[3], cm2[3], cf[3];              // -ad(V)^T (G V) = (w x Wm + v x Wf, w x Wf)
      cross3(Vsave[i],   W,   cm1);
      cross3(Vsave[i]+3, W+3, cm2);
      cross3(Vsave[i],   W+3, cf);
#pragma unroll
      for (int u=0;u<3;++u){
        F[u]   = Fn[u]   + U[u]   + cm1[u] + cm2[u];
        F[3+u] = Fn[3+u] + U[3+u] + cf[u];
      }
      h[i] = dot6(F, sA[i]);
    }

    // ---- mass matrix lower triangle via reduced ID (V==0, g=0, Ftip=0) ----
    float Mm[7][7];
#pragma unroll
    for (int j=0;j<7;++j){
      float Vdc[7][6];
#pragma unroll
      for (int u=0;u<6;++u) Vdc[j][u] = sA[j][u];
#pragma unroll
      for (int i2=0;i2<7;++i2){
        if (i2 <= j) continue;
        adTapply(myT + i2*12, myT + i2*12 + 9, Vdc[i2-1], Vdc[i2]);
      }
      float Fc[6] = {0,0,0,0,0,0};
#pragma unroll
      for (int i2=6;i2>=0;--i2){
        if (i2 < j) continue;
        if (i2 < 6) {
          float Ftmp[6];
          adTTapply(myT + (i2+1)*12, myT + (i2+1)*12 + 9, Fc, Ftmp);
#pragma unroll
          for (int u=0;u<6;++u) Fc[u] = Ftmp[u];
        }
        float U[6]; g6mv(sG[i2], Vdc[i2], U);
#pragma unroll
        for (int u=0;u<6;++u) Fc[u] += U[u];
        Mm[i2][j] = dot6(Fc, sA[i2]);
      }
    }

    // ---- solve Mm x = tf - h  (Cholesky, SPD, fully unrolled -> registers) ----
    float rhs[7];
#pragma unroll
    for (int j=0;j<7;++j) rhs[j] = tf[j] - h[j];
#pragma unroll
    for (int k=0;k<7;++k){
      float d = Mm[k][k];
#pragma unroll
      for (int tt=0;tt<7;++tt) if (tt<k) d -= Mm[k][tt]*Mm[k][tt];
      d = sqrtf(d);
      Mm[k][k] = d;
      float inv = 1.f/d;
#pragma unroll
      for (int r=0;r<7;++r){
        if (r<=k) continue;
        float e = Mm[r][k];
#pragma unroll
        for (int tt=0;tt<7;++tt) if (tt<k) e -= Mm[r][tt]*Mm[k][tt];
        Mm[r][k] = e*inv;
      }
    }
    float y[7];
#pragma unroll
    for (int i=0;i<7;++i){
      float s = rhs[i];
#pragma unroll
      for (int tt=0;tt<7;++tt) if (tt<i) s -= Mm[i][tt]*y[tt];
      y[i] = s / Mm[i][i];
    }
    float x[7];
#pragma unroll
    for (int i=6;i>=0;--i){
      float s = y[i];
#pragma unroll
      for (int tt=0;tt<7;++tt) if (tt>i) s -= Mm[tt][i]*x[tt];
      x[i] = s / Mm[i][i];
    }

    // ---- RK4 accumulate / next stage state ----
    float wgt = (st==0 || st==3) ? 1.f : 2.f;
#pragma unroll
    for (int j=0;j<7;++j){ accq[j] += wgt*dqs[j]; accdq[j] += wgt*x[j]; }
    if (st < 3) {
      float coef = (st<2) ? (0.5f*DT) : DT;
      float nq[7], nd[7];
#pragma unroll
      for (int j=0;j<7;++j){ nq[j] = q0[j] + coef*dqs[j]; nd[j] = dq0[j] + coef*x[j]; }
#pragma unroll
      for (int j=0;j<7;++j){ qs[j]=nq[j]; dqs[j]=nd[j]; }
    }
  }

  // ---- finalize: wrap q, clip dq ----
  const float PI = 3.14159265358979323846f;
  float qw[7], dqc[7];
#pragma unroll
  for (int j=0;j<7;++j){
    float s1q = q0[j]  + (DT/6.f)*accq[j];
    float s1d = dq0[j] + (DT/6.f)*accdq[j];
    float m = fmodf(s1q + PI, 2.f*PI);
    if (m < 0.f) m += 2.f*PI;            // jnp.mod semantics: result in [0, 2pi)
    qw[j]  = m - PI;
    dqc[j] = fminf(fmaxf(s1d, -MAX_VEL), MAX_VEL);
  }
#pragma unroll
  for (int j=0;j<7;++j){
    out_state[idx*14 + j]     = qw[j];
    out_state[idx*14 + 7 + j] = dqc[j];
  }

  // ---- FK: T = prod_i (M_i * exp(A_i q_i)), ee = (T * M_7).p[0:2] ----
  float Tr[9] = {1,0,0, 0,1,0, 0,0,1};
  float Tp[3] = {0,0,0};
#pragma unroll
  for (int i=0;i<7;++i){
    float R1[9]; mat3mul(Tr, sMr[i], R1);
    float p1[3]; mat3vec(Tr, sMp[i], p1);
#pragma unroll
    for (int u=0;u<3;++u) p1[u] += Tp[u];
    float w[3], v[3], Er[9], Ep[3];
#pragma unroll
    for (int u=0;u<3;++u){ w[u]=sA[i][u]*qw[i]; v[u]=sA[i][3+u]*qw[i]; }
    exp_se3(w, v, Er, Ep);
    float R2[9]; mat3mul(R1, Er, R2);
    float p2[3]; mat3vec(R1, Ep, p2);
#pragma unroll
    for (int u=0;u<9;++u) Tr[u] = R2[u];
#pragma unroll
    for (int u=0;u<3;++u) Tp[u] = p2[u] + p1[u];
  }
  float pe[3]; mat3vec(Tr, sMp[7], pe);
  out_ee[idx*2+0] = pe[0] + Tp[0];
  out_ee[idx*2+1] = pe[1] + Tp[1];
}

extern "C" void kernel_launch(void* const* d_in, const int* in_sizes, int n_in,
                              void* d_out, int out_size, void* d_ws, size_t ws_size,
                              hipStream_t stream) {
  (void)in_sizes; (void)n_in; (void)out_size; (void)d_ws; (void)ws_size;
  const float* state  = (const float*)d_in[0];
  const float* torque = (const float*)d_in[1];
  const float* M      = (const float*)d_in[2];
  const float* A      = (const float*)d_in[3];
  const float* L      = (const float*)d_in[4];
  const float* g      = (const float*)d_in[5];
  const float* ft     = (const float*)d_in[6];
  float* out       = (float*)d_out;
  float* out_state = out;                 // [B][14]
  float* out_ee    = out + BATCH*14;      // [B][2]
  dim3 grid(BATCH/NT), block(NT);
  arm_rk4_kernel<<<grid, block, 0, stream>>>(state, torque, M, A, L, g, ft,
                                             out_state, out_ee);
}